// HeteroGNN_24558622998759
// MI455X (gfx1250) — compile-verified
//
#include <hip/hip_runtime.h>
#include <cstddef>

#define HID   128
#define OUTD  64
#define NU    50000
#define NI    50000
#define NE    800000
#define SLOPE 0.2f

typedef float v2f __attribute__((ext_vector_type(2)));
typedef float v8f __attribute__((ext_vector_type(8)));

// ---------------- helpers ----------------
__device__ __forceinline__ float lrelu(float x) { return x > 0.f ? x : x * SLOPE; }

// order-preserving float <-> uint map for atomicMax on floats
__device__ __forceinline__ unsigned f2ord(float f) {
    unsigned b = __float_as_uint(f);
    return (b & 0x80000000u) ? ~b : (b | 0x80000000u);
}
__device__ __forceinline__ float ord2f(unsigned u) {
    unsigned b = (u & 0x80000000u) ? (u & 0x7FFFFFFFu) : ~u;
    return __uint_as_float(b);
}

__device__ __forceinline__ void atomAddF(float* p, float v) {
    // lowers to global_atomic_add_f32 (agent scope, relaxed)
    __hip_atomic_fetch_add(p, v, __ATOMIC_RELAXED, __HIP_MEMORY_SCOPE_AGENT);
}

// ---------------- WMMA fp32 GEMM: Y[N,M] = X[N,K] @ W[K,M] + bias ----------------
// One wave per 16x16 output tile; K-loop of v_wmma_f32_16x16x4_f32.
// N, M multiples of 16; K multiple of 4. EXEC all-ones per wave (uniform early-out).
__global__ void __launch_bounds__(256)
gemm_wmma_bias(const float* __restrict__ X, const float* __restrict__ W,
               const float* __restrict__ B, float* __restrict__ Y,
               int N, int K, int M) {
    int gid  = blockIdx.x * blockDim.x + threadIdx.x;
    int wave = gid >> 5;
    int lane = threadIdx.x & 31;
    int tilesM = M >> 4;
    int tiles  = (N >> 4) * tilesM;
    if (wave >= tiles) return;                    // wave-uniform
    int tr = wave / tilesM, tc = wave % tilesM;
    int row0 = tr << 4, col0 = tc << 4;
    int half = lane >> 4, idx = lane & 15;

    const float* xrow = X + (size_t)(row0 + idx) * K;   // A row for this lane
    const float* wcol = W + col0 + idx;                 // B column for this lane

    v8f c = {0.f,0.f,0.f,0.f,0.f,0.f,0.f,0.f};
    for (int k = 0; k < K; k += 4) {
        int ka = k + 2 * half;          // lanes 0-15: K=k,k+1 ; lanes 16-31: K=k+2,k+3
        v2f a, b;
        a.x = xrow[ka];
        a.y = xrow[ka + 1];
        b.x = wcol[(size_t)ka * M];
        b.y = wcol[(size_t)(ka + 1) * M];
        c = __builtin_amdgcn_wmma_f32_16x16x4_f32(false, a, false, b,
                                                  (short)0, c, false, false);
    }
    float bv = (B != nullptr) ? B[col0 + idx] : 0.f;
#pragma unroll
    for (int v = 0; v < 8; ++v) {       // D: vgpr v -> row v (lanes 0-15) / v+8 (16-31)
        int r = row0 + v + 8 * half;
        Y[(size_t)r * M + col0 + idx] = c[v] + bv;
    }
}

// ---------------- edge pass A: score + segment max ----------------
// one wave per edge; lane j handles dims 4j..4j+3 (float4, fully coalesced)
__global__ void __launch_bounds__(256)
edge_pass_a(const float* __restrict__ xl, const float* __restrict__ xr,
            const int* __restrict__ src, const int* __restrict__ dst,
            const float* __restrict__ ea, const float* __restrict__ We,
            const float* __restrict__ att,
            float* __restrict__ esc, unsigned* __restrict__ emax, int E) {
    int gw   = (blockIdx.x * blockDim.x + threadIdx.x) >> 5;
    int lane = threadIdx.x & 31;
    if (gw >= E) return;
    int s = src[gw], d = dst[gw];
    const float4 a = *(const float4*)(xl + (size_t)s * HID + lane * 4);
    const float4 b = *(const float4*)(xr + (size_t)d * HID + lane * 4);
    float m0 = a.x + b.x, m1 = a.y + b.y, m2 = a.z + b.z, m3 = a.w + b.w;
    if (ea != nullptr) {                       // fused lin_edge: ea[e] @ We
        const float* ep = ea + (size_t)gw * 16;
#pragma unroll
        for (int k = 0; k < 16; ++k) {
            float ev = ep[k];
            const float4 w = *(const float4*)(We + k * HID + lane * 4);
            m0 += ev * w.x; m1 += ev * w.y; m2 += ev * w.z; m3 += ev * w.w;
        }
    }
    const float4 at = *(const float4*)(att + lane * 4);
    float p = lrelu(m0) * at.x + lrelu(m1) * at.y + lrelu(m2) * at.z + lrelu(m3) * at.w;
#pragma unroll
    for (int off = 16; off > 0; off >>= 1) p += __shfl_xor(p, off, 32);
    if (lane == 0) {
        esc[gw] = p;
        atomicMax(emax + d, f2ord(p));         // global_atomic_max_u32
    }
}

// ---------------- edge pass B: w = exp(e - max), denom += w ----------------
__global__ void __launch_bounds__(256)
edge_pass_b(const int* __restrict__ dst, const unsigned* __restrict__ emax,
            float* __restrict__ esc, float* __restrict__ denom, int E) {
    int e = blockIdx.x * blockDim.x + threadIdx.x;
    if (e >= E) return;
    int d = dst[e];
    float w = __expf(esc[e] - ord2f(emax[d]));
    esc[e] = w;
    atomAddF(denom + d, w);
}

// ---------------- edge pass C: acc[dst] += alpha * xl[src] ----------------
__global__ void __launch_bounds__(256)
edge_pass_c(const float* __restrict__ xl,
            const int* __restrict__ src, const int* __restrict__ dst,
            const float* __restrict__ esc, const float* __restrict__ denom,
            float* __restrict__ acc, int E) {
    int gw   = (blockIdx.x * blockDim.x + threadIdx.x) >> 5;
    int lane = threadIdx.x & 31;
    if (gw >= E) return;
    int s = src[gw], d = dst[gw];
    float alpha = esc[gw] / (denom[d] + 1e-16f);
    const float4 a = *(const float4*)(xl + (size_t)s * HID + lane * 4);
    float* o = acc + (size_t)d * HID + lane * 4;
    atomAddF(o + 0, alpha * a.x);
    atomAddF(o + 1, alpha * a.y);
    atomAddF(o + 2, alpha * a.z);
    atomAddF(o + 3, alpha * a.w);
}

// ---------------- elementwise kernels ----------------
__global__ void zero_u32(unsigned* p, int n) {
    int i = blockIdx.x * blockDim.x + threadIdx.x;
    if (i < n) p[i] = 0u;
}
__global__ void add_bias_vec(float* __restrict__ acc, const float* __restrict__ bias, int nodes) {
    int i = blockIdx.x * blockDim.x + threadIdx.x;
    if (i < nodes * HID) acc[i] += bias[i & (HID - 1)];
}
__global__ void relu_copy(float* __restrict__ dstp, const float* __restrict__ srcp, int n) {
    int i = blockIdx.x * blockDim.x + threadIdx.x;
    if (i < n) dstp[i] = fmaxf(srcp[i], 0.f);
}

// ---------------- host-side orchestration ----------------
struct GatP { const float *Wl, *bl, *Wr, *br, *att, *bias, *We; };

static inline dim3 blocks_for(long long threads) {
    return dim3((unsigned)((threads + 255) / 256));
}

static void launch_gemm(const float* X, const float* W, const float* B, float* Y,
                        int N, int K, int M, hipStream_t st) {
    long long waves = (long long)(N >> 4) * (M >> 4);
    gemm_wmma_bias<<<blocks_for(waves * 32), 256, 0, st>>>(X, W, B, Y, N, K, M);
}

static void run_gat(const GatP& g,
                    const float* xsrc, int n_src, int k_src,
                    const float* xdst, int n_dst, int k_dst,
                    const int* ei, const float* ea,
                    float* xl, float* xr, float* esc, float* denom, unsigned* emax,
                    float* acc, hipStream_t st) {
    const int* src = ei;
    const int* dst = ei + NE;
    // xl = x_src @ Wl + bl ; xr = x_dst @ Wr + br   (WMMA fp32)
    launch_gemm(xsrc, g.Wl, g.bl, xl, n_src, k_src, HID, st);
    launch_gemm(xdst, g.Wr, g.br, xr, n_dst, k_dst, HID, st);
    // reset segment state
    zero_u32<<<blocks_for(n_dst), 256, 0, st>>>(emax, n_dst);
    zero_u32<<<blocks_for(n_dst), 256, 0, st>>>((unsigned*)denom, n_dst);
    // scatter-softmax passes
    edge_pass_a<<<blocks_for((long long)NE * 32), 256, 0, st>>>(
        xl, xr, src, dst, ea, g.We, g.att, esc, emax, NE);
    edge_pass_b<<<blocks_for(NE), 256, 0, st>>>(dst, emax, esc, denom, NE);
    edge_pass_c<<<blocks_for((long long)NE * 32), 256, 0, st>>>(
        xl, src, dst, esc, denom, acc, NE);
    // + per-relation bias (hetero 'sum' aggregation accumulates into acc)
    add_bias_vec<<<blocks_for((long long)n_dst * HID), 256, 0, st>>>(acc, g.bias, n_dst);
}

extern "C" void kernel_launch(void* const* d_in, const int* in_sizes, int n_in,
                              void* d_out, int out_size, void* d_ws, size_t ws_size,
                              hipStream_t stream) {
    (void)in_sizes; (void)n_in; (void)out_size; (void)ws_size;
    const float* x_user = (const float*)d_in[0];
    const float* x_item = (const float*)d_in[1];
    const int*   ei0    = (const int*)d_in[2];
    const int*   ei1    = (const int*)d_in[3];
    const int*   ei2    = (const int*)d_in[4];
    const float* ea1    = (const float*)d_in[5];
    const float* ea2    = (const float*)d_in[6];

    // params flattened in dict-insertion order: per GAT = Wl, bl, Wr, br, att, bias, [We]
    int p = 7;
    auto nextGat = [&](bool hasWe) {
        GatP g;
        g.Wl   = (const float*)d_in[p++];
        g.bl   = (const float*)d_in[p++];
        g.Wr   = (const float*)d_in[p++];
        g.br   = (const float*)d_in[p++];
        g.att  = (const float*)d_in[p++];
        g.bias = (const float*)d_in[p++];
        g.We   = hasWe ? (const float*)d_in[p++] : nullptr;
        return g;
    };
    GatP l0r0 = nextGat(false), l0r1 = nextGat(true), l0r2 = nextGat(true);
    GatP l1r0 = nextGat(false), l1r1 = nextGat(true), l1r2 = nextGat(true);
    const float* linWu = (const float*)d_in[p++];
    const float* linbu = (const float*)d_in[p++];
    const float* linWi = (const float*)d_in[p++];
    const float* linbi = (const float*)d_in[p++];

    // workspace layout
    const size_t NODE = (size_t)NU * HID;   // == NI*HID
    float* ws   = (float*)d_ws;
    float* f_u   = ws;              // layer features (user)
    float* f_i   = f_u + NODE;      // layer features (item)
    float* acc_u = f_i + NODE;      // hetero-sum accumulator (user)
    float* acc_i = acc_u + NODE;    // hetero-sum accumulator (item)
    float* xl    = acc_i + NODE;    // per-relation transformed src feats
    float* xr    = xl + NODE;       // per-relation transformed dst feats
    float* esc   = xr + NODE;       // per-edge score / weight   [NE]
    float* denom = esc + NE;        // segment softmax denom     [max nodes]
    unsigned* emax = (unsigned*)(denom + NU);  // segment max (ordered-uint) [max nodes]

    float* out_u = (float*)d_out;
    float* out_i = out_u + (size_t)NU * OUTD;

    // ---------------- layer 0 (inputs: x_user[*,64], x_item[*,32]) ----------------
    zero_u32<<<blocks_for((long long)NODE), 256, 0, stream>>>((unsigned*)acc_u, (int)NODE);
    zero_u32<<<blocks_for((long long)NODE), 256, 0, stream>>>((unsigned*)acc_i, (int)NODE);
    // r0: user -> user (no edge attr)
    run_gat(l0r0, x_user, NU, 64, x_user, NU, 64, ei0, nullptr,
            xl, xr, esc, denom, emax, acc_u, stream);
    // r2: item -> user (ea2)
    run_gat(l0r2, x_item, NI, 32, x_user, NU, 64, ei2, ea2,
            xl, xr, esc, denom, emax, acc_u, stream);
    // r1: user -> item (ea1)
    run_gat(l0r1, x_user, NU, 64, x_item, NI, 32, ei1, ea1,
            xl, xr, esc, denom, emax, acc_i, stream);
    relu_copy<<<blocks_for((long long)NODE), 256, 0, stream>>>(f_u, acc_u, (int)NODE);
    relu_copy<<<blocks_for((long long)NODE), 256, 0, stream>>>(f_i, acc_i, (int)NODE);

    // ---------------- layer 1 (inputs: f_u[*,128], f_i[*,128]) ----------------
    zero_u32<<<blocks_for((long long)NODE), 256, 0, stream>>>((unsigned*)acc_u, (int)NODE);
    zero_u32<<<blocks_for((long long)NODE), 256, 0, stream>>>((unsigned*)acc_i, (int)NODE);
    run_gat(l1r0, f_u, NU, HID, f_u, NU, HID, ei0, nullptr,
            xl, xr, esc, denom, emax, acc_u, stream);
    run_gat(l1r2, f_i, NI, HID, f_u, NU, HID, ei2, ea2,
            xl, xr, esc, denom, emax, acc_u, stream);
    run_gat(l1r1, f_u, NU, HID, f_i, NI, HID, ei1, ea1,
            xl, xr, esc, denom, emax, acc_i, stream);
    relu_copy<<<blocks_for((long long)NODE), 256, 0, stream>>>(f_u, acc_u, (int)NODE);
    relu_copy<<<blocks_for((long long)NODE), 256, 0, stream>>>(f_i, acc_i, (int)NODE);

    // ---------------- final linear projections (WMMA fp32) ----------------
    launch_gemm(f_u, linWu, linbu, out_u, NU, HID, OUTD, stream);
    launch_gemm(f_i, linWi, linbi, out_i, NI, HID, OUTD, stream);
}